// ElementLinear_35777077575975
// MI455X (gfx1250) — compile-verified
//
#include <hip/hip_runtime.h>

typedef float v2f __attribute__((ext_vector_type(2)));
typedef float v8f __attribute__((ext_vector_type(8)));

#define IN_DIM  128
#define OUT_DIM 128
#define R_DIM   16
#define ALPHA_F 0.0883883476483184f   // 1/sqrt(128)

// One workgroup (8 waves, 256 threads) per batch element b.
// Wave w computes the 16x16 output tile rows [w*16, w*16+16) x cols [0,16).
__global__ __launch_bounds__(256)
void element_linear_wmma_f32(const float* __restrict__ T,
                             const float* __restrict__ attrs,
                             const float* __restrict__ W,
                             const float* __restrict__ bias,
                             float* __restrict__ out,
                             int E) {
    __shared__ float sT[IN_DIM * R_DIM];   // 8 KB: T[b] tile, [K=128][N=16]
    __shared__ int   sIdx;

    const int b    = blockIdx.x;
    const int tid  = threadIdx.x;
    const int wave = tid >> 5;
    const int lane = tid & 31;
    const int n    = lane & 15;     // column within 16-wide tile / M row within A
    const int lh   = lane >> 4;     // lane half selects K sub-pair

    // ---- Stage T[b] (2048 floats) into LDS with coalesced 128b loads ----
    const float4* gT = (const float4*)(T + (size_t)b * (IN_DIM * R_DIM));
    float4* s4 = (float4*)sT;
    s4[tid]       = gT[tid];
    s4[tid + 256] = gT[tid + 256];

    // ---- argmax(node_attrs[b]) , first-max tie-break like jnp.argmax ----
    if (tid == 0) {
        const float* a = attrs + (size_t)b * E;
        float best = a[0]; int bi = 0;
        for (int e = 1; e < E; ++e) {
            float v = a[e];
            if (v > best) { best = v; bi = e; }
        }
        sIdx = bi;
    }
    __syncthreads();

    const int idx = sIdx;

    // A-matrix 16x4 f32 layout: lane holds row M = lane%16,
    // VGPR v holds K = k0 + (lane>>4)*2 + v  -> contiguous float2 per lane.
    const float* Arow = W + (size_t)idx * (OUT_DIM * IN_DIM)
                          + (size_t)(wave * 16 + n) * IN_DIM + lh * 2;

    // B-matrix 4x16 f32 layout: lane holds col N = lane%16,
    // VGPR v holds K = k0 + (lane>>4)*2 + v, read from LDS (stride 16 floats).
    const float* Bcol = sT + n + (lh * 2) * R_DIM;

    v8f acc = {0.f, 0.f, 0.f, 0.f, 0.f, 0.f, 0.f, 0.f};

    #pragma unroll
    for (int k0 = 0; k0 < IN_DIM; k0 += 4) {
        v2f av = *(const v2f*)(Arow + k0);          // W[M, k0+2*lh .. +1]
        v2f bv;
        bv.x = Bcol[k0 * R_DIM];                    // T[k0+2*lh  , n]
        bv.y = Bcol[k0 * R_DIM + R_DIM];            // T[k0+2*lh+1, n]
        // D = A x B + C  (16x16x4, f32)
        acc = __builtin_amdgcn_wmma_f32_16x16x4_f32(
                  /*neg_a=*/false, av,
                  /*neg_b=*/false, bv,
                  /*c_mod=*/(short)0, acc,
                  /*reuse_a=*/false, /*reuse_b=*/false);
    }

    // ---- Epilogue: out = acc*ALPHA + bias[idx], C/D layout:
    //      VGPR v -> M = v + lh*8, N = lane%16
    const float* bp = bias + (size_t)idx * OUT_DIM + wave * 16 + lh * 8;
    float* op = out + (size_t)b * (OUT_DIM * R_DIM)
                    + (size_t)(wave * 16 + lh * 8) * R_DIM + n;
    #pragma unroll
    for (int v = 0; v < 8; ++v) {
        op[v * R_DIM] = fmaf(acc[v], ALPHA_F, bp[v]);
    }
}

extern "C" void kernel_launch(void* const* d_in, const int* in_sizes, int n_in,
                              void* d_out, int out_size, void* d_ws, size_t ws_size,
                              hipStream_t stream) {
    const float* T     = (const float*)d_in[0];   // [B,128,16]
    const float* attrs = (const float*)d_in[1];   // [B,E]
    const float* W     = (const float*)d_in[2];   // [E,128,128]
    const float* bias  = (const float*)d_in[3];   // [E,128]
    float* out         = (float*)d_out;           // [B,128,16]

    const int B = in_sizes[0] / (IN_DIM * R_DIM);
    const int E = (B > 0) ? (in_sizes[1] / B) : 10;

    dim3 grid(B), block(256);
    hipLaunchKernelGGL(element_linear_wmma_f32, grid, block, 0, stream,
                       T, attrs, W, bias, out, E);
}